// SinkhornDistance_17111149707520
// MI455X (gfx1250) — compile-verified
//
#include <hip/hip_runtime.h>
#include <math.h>

// ---------------- problem constants ----------------
#define EPSV     0.1f
#define INV_EPS  10.0f
#define THRESH   0.1f
#define MAX_ITER 100
#define NB  128          // batch
#define P   512          // points per set
#define PP  (P * P)      // 262144
// log(1/512 + 1e-8)
#define LOG_MU  (-6.2383195f)
#define LOG_NU  (-6.2383195f)

typedef __attribute__((ext_vector_type(2))) float v2f;
typedef __attribute__((ext_vector_type(8))) float v8f;

// ---------------- workspace layout (floats) ----------------
// W   : 512x512 weight |1/(S S^T) + 1e-5|        [0, 262144)
// u   : 128x512                                   [262144, 327680)
// v   : 128x512                                   [327680, 393216)
// du  : 128x512 per-row |u_new - u|               [393216, 458752)
// done: 1 int                                     [458752]
#define WS_W    0
#define WS_U    (WS_W + PP)
#define WS_V    (WS_U + NB * P)
#define WS_DU   (WS_V + NB * P)
#define WS_DONE (WS_DU + NB * P)

// ---------------- init: zero u, v, du, done, cost ----------------
__global__ void sk_init_kernel(float* __restrict__ ws, float* __restrict__ cost) {
  int idx = blockIdx.x * blockDim.x + threadIdx.x;
  if (idx < 3 * NB * P) ws[WS_U + idx] = 0.0f;
  if (idx < NB) cost[idx] = 0.0f;
  if (idx == 0) *(int*)(ws + WS_DONE) = 0;
}

// ---------------- Gram via WMMA: W = |1/(species species^T) + 1e-5| ------
// one wave32 per 16x16 tile; K = 8 -> two chained V_WMMA_F32_16X16X4_F32
__global__ void sk_gram_w_kernel(const float* __restrict__ sp, float* __restrict__ W) {
  const int tile = blockIdx.x;        // 0..1023
  const int mT = tile >> 5;
  const int nT = tile & 31;
  const int lane = threadIdx.x;       // 0..31
  const int half = lane >> 4;
  const int l = lane & 15;
  const int rowA = mT * 16 + l;
  const int colB = nT * 16 + l;
  const int k0 = 2 * half;            // VGPR0 holds K=2*half, VGPR1 holds K=2*half+1

  v2f a0, a1, b0, b1;
  a0.x = sp[rowA * 8 + k0];      a0.y = sp[rowA * 8 + k0 + 1];
  a1.x = sp[rowA * 8 + 4 + k0];  a1.y = sp[rowA * 8 + 4 + k0 + 1];
  // B is 4x16 = (species^T) tile: B[k][n] = species[n][k] -> same pattern as A
  b0.x = sp[colB * 8 + k0];      b0.y = sp[colB * 8 + k0 + 1];
  b1.x = sp[colB * 8 + 4 + k0];  b1.y = sp[colB * 8 + 4 + k0 + 1];

  v8f c = {0.f, 0.f, 0.f, 0.f, 0.f, 0.f, 0.f, 0.f};
  c = __builtin_amdgcn_wmma_f32_16x16x4_f32(false, a0, false, b0, (short)0, c, false, false);
  c = __builtin_amdgcn_wmma_f32_16x16x4_f32(false, a1, false, b1, (short)0, c, false, false);

#pragma unroll
  for (int r = 0; r < 8; ++r) {
    const int M = r + 8 * half;     // C/D layout: lanes 16-31 hold rows 8..15
    const int Nc = l;
    const float s = c[r];
    W[(mT * 16 + M) * P + nT * 16 + Nc] = fabsf(1.0f / s + 1e-5f);
  }
}

// ---------------- cost matrix: dist_sqr and C ----------------
__global__ void sk_cost_kernel(const float* __restrict__ x, const float* __restrict__ y,
                               const float* __restrict__ W, float* __restrict__ Cout,
                               float* __restrict__ Dout) {
  const int j = blockIdx.x * 256 + threadIdx.x;
  const int i = blockIdx.y;
  const int n = blockIdx.z;
  const float x0 = x[(n * P + i) * 3 + 0];
  const float x1 = x[(n * P + i) * 3 + 1];
  const float x2 = x[(n * P + i) * 3 + 2];
  const float y0 = y[(n * P + j) * 3 + 0];
  const float y1 = y[(n * P + j) * 3 + 1];
  const float y2 = y[(n * P + j) * 3 + 2];
  const float dx0 = x0 - y0, dx1 = x1 - y1, dx2 = x2 - y2;
  const float p0 = rintf(dx0) - dx0;  // round-to-nearest-even, like jnp.round
  const float p1 = rintf(dx1) - dx1;
  const float p2 = rintf(dx2) - dx2;
  const float d2 = p0 * p0 + p1 * p1 + p2 * p2;
  const size_t idx = ((size_t)(n * P + i)) * P + j;
  Dout[idx] = d2;
  Cout[idx] = d2 * W[i * P + j];
}

// ---------------- Sinkhorn U step: row logsumexp, in-place u, du out -----
// one wave per row i; block = 8 waves; v[n,:] staged in LDS
__global__ void sk_u_kernel(const float* __restrict__ C, float* __restrict__ u,
                            const float* __restrict__ v, float* __restrict__ du,
                            const int* __restrict__ done) {
  if (*done) return;
  __shared__ float vs[P];
  const int n = blockIdx.x >> 6;           // 64 blocks per batch element
  const int rowg = blockIdx.x & 63;
  const int tid = threadIdx.x;
  for (int t = tid; t < P; t += 256) vs[t] = v[n * P + t];
  __syncthreads();

  const int wave = tid >> 5;
  const int lane = tid & 31;
  const int i = rowg * 8 + wave;
  const float* Crow = C + ((size_t)(n * P + i)) * P;
  const float ui = u[n * P + i];

  float t[16];
  float m = -INFINITY;
#pragma unroll
  for (int k = 0; k < 16; ++k) {
    const int j = lane + 32 * k;
    t[k] = (ui + vs[j] - Crow[j]) * INV_EPS;
    m = fmaxf(m, t[k]);
  }
  for (int off = 16; off > 0; off >>= 1) m = fmaxf(m, __shfl_xor(m, off, 32));
  float s = 0.0f;
#pragma unroll
  for (int k = 0; k < 16; ++k) s += __expf(t[k] - m);
  for (int off = 16; off > 0; off >>= 1) s += __shfl_xor(s, off, 32);

  const float lse = m + __logf(s);
  const float unew = EPSV * (LOG_MU - lse) + ui;
  if (lane == 0) {
    u[n * P + i] = unew;
    du[n * P + i] = fabsf(unew - ui);
  }
}

// ---------------- Sinkhorn V step: column logsumexp (online), in-place ---
// block = 512 threads = one batch element; thread j streams down column j
__global__ void sk_v_kernel(const float* __restrict__ C, const float* __restrict__ u,
                            float* __restrict__ v, const int* __restrict__ done) {
  if (*done) return;
  __shared__ float us[P];
  const int n = blockIdx.x;
  const int j = threadIdx.x;
  us[j] = u[n * P + j];
  __syncthreads();
  const float* Cn = C + (size_t)n * PP;
  const float vj = v[n * P + j];
  float m = -INFINITY, s = 0.0f;
  for (int i = 0; i < P; ++i) {
    const float tt = (us[i] + vj - Cn[(size_t)i * P + j]) * INV_EPS;
    if (tt <= m) {
      s += __expf(tt - m);
    } else {
      s = s * __expf(m - tt) + 1.0f;
      m = tt;
    }
  }
  const float lse = m + __logf(s);
  v[n * P + j] = EPSV * (LOG_NU - lse) + vj;
}

// ---------------- err = mean_n sum_i |du|; set done (deterministic) ------
__global__ void sk_err_kernel(const float* __restrict__ du, int* __restrict__ done) {
  if (*done) return;
  __shared__ float sh[256];
  const int tid = threadIdx.x;
  float s = 0.0f;
  for (int k = tid; k < NB * P; k += 256) s += du[k];
  sh[tid] = s;
  __syncthreads();
  for (int off = 128; off > 0; off >>= 1) {
    if (tid < off) sh[tid] += sh[tid + off];
    __syncthreads();
  }
  if (tid == 0 && (sh[0] * (1.0f / (float)NB)) < THRESH) *done = 1;
}

// ---------------- pi = exp(M(u,v)) ----------------
__global__ void sk_pi_kernel(const float* __restrict__ C, const float* __restrict__ u,
                             const float* __restrict__ v, float* __restrict__ pi) {
  const int j = blockIdx.x * 256 + threadIdx.x;
  const int i = blockIdx.y;
  const int n = blockIdx.z;
  const size_t idx = ((size_t)(n * P + i)) * P + j;
  pi[idx] = __expf((u[n * P + i] + v[n * P + j] - C[idx]) * INV_EPS);
}

// ---------------- cost[n] = sum(pi*C), fixed-order reduction -------------
__global__ void sk_costsum_kernel(const float* __restrict__ pi, const float* __restrict__ C,
                                  float* __restrict__ cost) {
  __shared__ float sh[256];
  const int n = blockIdx.x;
  const int tid = threadIdx.x;
  const size_t base = (size_t)n * PP;
  float s = 0.0f;
  for (int k = tid; k < PP; k += 256) s += pi[base + k] * C[base + k];
  sh[tid] = s;
  __syncthreads();
  for (int off = 128; off > 0; off >>= 1) {
    if (tid < off) sh[tid] += sh[tid + off];
    __syncthreads();
  }
  if (tid == 0) cost[n] = sh[0];
}

// ---------------- host launcher ----------------
extern "C" void kernel_launch(void* const* d_in, const int* in_sizes, int n_in,
                              void* d_out, int out_size, void* d_ws, size_t ws_size,
                              hipStream_t stream) {
  const float* x  = (const float*)d_in[0];   // (128, 512, 3)
  const float* y  = (const float*)d_in[1];   // (128, 512, 3)
  const float* sp = (const float*)d_in[2];   // (512, 8)

  float* out = (float*)d_out;
  const size_t NPP = (size_t)NB * PP;        // 33554432
  float* cost_out = out;                     // (128,)
  float* pi_out   = out + NB;                // (128,512,512)
  float* C_out    = pi_out + NPP;            // (128,512,512)
  float* D_out    = C_out + NPP;             // (128,512,512)

  float* ws   = (float*)d_ws;
  float* W    = ws + WS_W;
  float* u    = ws + WS_U;
  float* v    = ws + WS_V;
  float* du   = ws + WS_DU;
  int*   done = (int*)(ws + WS_DONE);

  // 1. zero u, v, du, done, cost
  sk_init_kernel<<<(3 * NB * P + 255) / 256, 256, 0, stream>>>(ws, cost_out);

  // 2. Gram matrix weight via WMMA (32x32 tiles of 16x16, one wave each)
  sk_gram_w_kernel<<<1024, 32, 0, stream>>>(sp, W);

  // 3. cost matrix + dist_sqr straight into d_out (C stays L2-resident)
  {
    dim3 grid(P / 256, P, NB);
    sk_cost_kernel<<<grid, 256, 0, stream>>>(x, y, W, C_out, D_out);
  }

  // 4. Sinkhorn iterations with reference-exact early-stop semantics
  for (int it = 0; it < MAX_ITER; ++it) {
    sk_u_kernel<<<NB * (P / 8), 256, 0, stream>>>(C_out, u, v, du, done);
    sk_v_kernel<<<NB, P, 0, stream>>>(C_out, u, v, done);
    sk_err_kernel<<<1, 256, 0, stream>>>(du, done);
  }

  // 5. transport plan
  {
    dim3 grid(P / 256, P, NB);
    sk_pi_kernel<<<grid, 256, 0, stream>>>(C_out, u, v, pi_out);
  }

  // 6. per-batch cost, deterministic fixed-order reduction
  sk_costsum_kernel<<<NB, 256, 0, stream>>>(pi_out, C_out, cost_out);
}